// FastDiffMPO_33964601376807
// MI455X (gfx1250) — compile-verified
//
#include <hip/hip_runtime.h>
#include <hip/hip_bf16.h>
#include <math.h>

// Problem constants (from reference)
#define BDIM 128
#define HDIM 32
#define NDIM 128
#define BH   (BDIM*HDIM)          // 4096
#define GAMMA_C 5.0f
#define COST_COEFF_C 1e-3f
#define KAPPA_C 2.0627128075074256f
#define ETA_FWD_C 0.05f
#define MAX_ITERS_C 300
#define TOL2_C 1e-12f             // TOL^2

typedef float v2f __attribute__((ext_vector_type(2)));
typedef float v8f __attribute__((ext_vector_type(8)));

// ---------------------------------------------------------------------------
// Stage 1: M[bh] = L[bh] * L[bh]^T  (4096 batched 128x128x128 f32 GEMMs)
// One workgroup (8 wave32) per (b,h). Uses V_WMMA_F32_16X16X4_F32.
// LDS stages L in 128x32 K-slices, pitch 36 words (bank-conflict free for
// both A-fragment reads (row wid*16+m) and B-fragment reads (row nt*16+m);
// lane halves land on disjoint bank sets).
// ---------------------------------------------------------------------------
#define KSLICE 32
#define NPIT   36   // 128*36*4B = 18KB LDS

__global__ void __launch_bounds__(256)
llt_gemm_kernel(const float* __restrict__ L, float* __restrict__ M)
{
    const int bh   = blockIdx.x;
    const int tid  = threadIdx.x;     // 0..255
    const int wid  = tid >> 5;        // wave id 0..7 -> row tile
    const int lane = tid & 31;
    const int lm   = lane & 15;       // M/N within tile
    const int kb   = (lane >> 4) * 2; // K base: lanes 0-15 -> K0/K1, 16-31 -> K2/K3

    __shared__ float Ls[NDIM * NPIT];

    const float* __restrict__ Lbh = L + (size_t)bh * (NDIM * NDIM);

    v8f acc[8];
#pragma unroll
    for (int i = 0; i < 8; ++i) acc[i] = (v8f){0.f,0.f,0.f,0.f,0.f,0.f,0.f,0.f};

    for (int ks = 0; ks < NDIM / KSLICE; ++ks) {
        const int k0 = ks * KSLICE;
        __syncthreads();
        // stage 128 rows x 32 cols of L into LDS (float4, coalesced)
        for (int idx = tid; idx < NDIM * (KSLICE / 4); idx += 256) {
            const int row = idx >> 3;           // 8 float4 per row-slice
            const int q   = idx & 7;
            const float4 val = *(const float4*)(Lbh + row * NDIM + k0 + q * 4);
            *(float4*)(&Ls[row * NPIT + q * 4]) = val;
        }
        __syncthreads();

#pragma unroll
        for (int kk = 0; kk < KSLICE; kk += 4) {
            // A fragment: A[m][k] = L[wid*16+m][k0+kk+kb..+1]
            const int arow = wid * 16 + lm;
            v2f a;
            a.x = Ls[arow * NPIT + kk + kb];
            a.y = Ls[arow * NPIT + kk + kb + 1];
#pragma unroll
            for (int nt = 0; nt < 8; ++nt) {
                // B fragment: B[k][n] = (L^T)[k][n] = L[nt*16+n][k0+kk+kb..+1]
                const int brow = nt * 16 + lm;
                v2f bf;
                bf.x = Ls[brow * NPIT + kk + kb];
                bf.y = Ls[brow * NPIT + kk + kb + 1];
                acc[nt] = __builtin_amdgcn_wmma_f32_16x16x4_f32(
                    false, a, false, bf, (short)0, acc[nt], false, false);
            }
        }
    }

    // write D tiles: VGPR r -> row (wid*16 + (lane>>4)*8 + r), col nt*16 + lm
    float* __restrict__ Mbh = M + (size_t)bh * (NDIM * NDIM);
    const int rowbase = wid * 16 + (lane >> 4) * 8;
#pragma unroll
    for (int nt = 0; nt < 8; ++nt) {
#pragma unroll
        for (int r = 0; r < 8; ++r) {
            Mbh[(size_t)(rowbase + r) * NDIM + nt * 16 + lm] = acc[nt][r];
        }
    }
}

// ---------------------------------------------------------------------------
// Deterministic block reductions (128 threads)
// ---------------------------------------------------------------------------
__device__ __forceinline__ float block_sum128(float v, float* red, int tid)
{
    red[tid] = v; __syncthreads();
#pragma unroll
    for (int s = 64; s > 0; s >>= 1) {
        if (tid < s) red[tid] += red[tid + s];
        __syncthreads();
    }
    float r = red[0]; __syncthreads();
    return r;
}

__device__ __forceinline__ float block_max128(float v, float* red, int tid)
{
    red[tid] = v; __syncthreads();
#pragma unroll
    for (int s = 64; s > 0; s >>= 1) {
        if (tid < s) red[tid] = fmaxf(red[tid], red[tid + s]);
        __syncthreads();
    }
    float r = red[0]; __syncthreads();
    return r;
}

// fused dual-sum reduction: returns (sum of a) in out[0], (sum of b) in out[1]
__device__ __forceinline__ void block_sum128x2(float a, float b, float* red,
                                               int tid, float* out)
{
    red[tid] = a; red[128 + tid] = b; __syncthreads();
#pragma unroll
    for (int s = 64; s > 0; s >>= 1) {
        if (tid < s) {
            red[tid]       += red[tid + s];
            red[128 + tid] += red[128 + tid + s];
        }
        __syncthreads();
    }
    out[0] = red[0]; out[1] = red[128]; __syncthreads();
}

// ---------------------------------------------------------------------------
// Stage 2: one mirror-descent step, with the global convergence check fused
// in via a last-block-done protocol (deterministic fixed-order reduction).
// One block (128 threads) per (b,h).
// v = M w (M symmetric -> thread tid reads column tid, fully coalesced).
// If *flag (converged): pass-through copy keeps ping-pong parity.
// ---------------------------------------------------------------------------
__global__ void __launch_bounds__(128)
md_step_kernel(const float* __restrict__ M,
               const float* __restrict__ mu,
               const float* __restrict__ w_prev,
               const float* __restrict__ cvar_limit,
               const float* __restrict__ win,
               float* __restrict__ wout,
               float* __restrict__ partials,
               int* __restrict__ flag,
               int* __restrict__ counter)
{
    const int bh  = blockIdx.x;
    const int b   = bh >> 5;   // H = 32
    const int h   = bh & 31;
    const int tid = threadIdx.x;

    __shared__ float wsh[NDIM];
    __shared__ float red[2 * NDIM];
    __shared__ int   isLast;

    const size_t rowoff = (size_t)bh * NDIM;
    const int converged = *flag;   // block-uniform

    if (converged) {
        // carry w forward unchanged (preserves ping-pong parity)
        wout[rowoff + tid] = win[rowoff + tid];
        if (tid == 0) partials[bh] = 0.f;
    } else {
        const float wi = win[rowoff + tid];
        wsh[tid] = wi;
        __syncthreads();

        // v = M w   (M symmetric: v_i = sum_j M[j][i]*w_j, coalesced in i=tid)
        const float* __restrict__ Mbh = M + (size_t)bh * (NDIM * NDIM);
        float v = 0.f;
#pragma unroll 8
        for (int j = 0; j < NDIM; ++j) {
            v += Mbh[(size_t)j * NDIM + tid] * wsh[j];
        }

        const float mui = mu[rowoff + tid];
        float sums[2];
        block_sum128x2(mui * wi, wi * v, red, tid, sums);   // mu_p, sigma^2
        const float mu_p  = sums[0];
        const float sigma = sqrtf(fmaxf(sums[1], 0.f));
        const float inv_sigma = 1.f / fmaxf(sigma, 1e-20f);

        const float viol = -mu_p + KAPPA_C * sigma - cvar_limit[b];
        // d/dv [100*softplus(50 v)/50] = 100*sigmoid(50 v)
        const float s = 100.f / (1.f + __expf(-50.f * viol));

        float g = -mui + 2.f * GAMMA_C * v + s * (-mui + KAPPA_C * v * inv_sigma);

        // trading-cost term: diff_h = w[h] - (h==0 ? w_prev : w[h-1])
        const float wm = (h == 0) ? w_prev[(size_t)b * NDIM + tid]
                                  : win[rowoff - NDIM + tid];
        const float d0 = wi - wm;
        g += COST_COEFF_C * d0 * rsqrtf(d0 * d0 + 1e-8f);
        if (h < HDIM - 1) {
            const float wp = win[rowoff + NDIM + tid];
            const float d1 = wp - wi;
            g -= COST_COEFF_C * d1 * rsqrtf(d1 * d1 + 1e-8f);
        }

        // mirror-descent update: softmax(log(w+1e-10) - eta*g)
        const float z = __logf(wi + 1e-10f) - ETA_FWD_C * g;
        const float zmax = block_max128(z, red, tid);
        const float e = __expf(z - zmax);
        const float esum = block_sum128(e, red, tid);
        const float wn = e / esum;

        wout[rowoff + tid] = wn;

        const float dd = wn - wi;
        const float ds = block_sum128(dd * dd, red, tid);
        if (tid == 0) partials[bh] = ds;
    }

    // ---- last-block-done convergence check (deterministic) ----
    __threadfence();                       // make partials[bh] device-visible
    if (tid == 0) {
        const int old = atomicAdd(counter, 1);
        isLast = (old == (int)gridDim.x - 1) ? 1 : 0;
    }
    __syncthreads();
    if (isLast) {
        __threadfence();                   // acquire side
        const volatile float* p = (const volatile float*)partials;
        float sacc = 0.f;
        for (int i = tid; i < BH; i += 128) sacc += p[i];   // fixed order
        const float dist2 = block_sum128(sacc, red, tid);
        if (tid == 0) {
            if (dist2 < TOL2_C) *flag = 1; // sticky; never cleared mid-solve
            *counter = 0;                  // rearm for next iteration
        }
    }
}

// ---------------------------------------------------------------------------
// Init: w0 = 1/N everywhere (in d_out = buffer A), flag = 0, counter = 0.
// ---------------------------------------------------------------------------
__global__ void __launch_bounds__(256)
init_kernel(float* __restrict__ w0, int* __restrict__ flag,
            int* __restrict__ counter)
{
    const int i = blockIdx.x * 256 + threadIdx.x;
    if (i < BH * NDIM) w0[i] = 1.f / (float)NDIM;
    if (i == 0) { *flag = 0; *counter = 0; }
}

// ---------------------------------------------------------------------------
// Host launcher. Inputs: mu (B,H,N) f32; L (B,H,N,N) f32; w_prev (B,N) f32;
// cvar_limit (B,) f32. Output: w (B,H,N) f32 in d_out.
// Workspace: M (256MB) | wB (2MB) | partials (16KB) | flag (4B) | counter (4B)
// ---------------------------------------------------------------------------
extern "C" void kernel_launch(void* const* d_in, const int* in_sizes, int n_in,
                              void* d_out, int out_size, void* d_ws, size_t ws_size,
                              hipStream_t stream)
{
    const float* mu         = (const float*)d_in[0];
    const float* L          = (const float*)d_in[1];
    const float* w_prev     = (const float*)d_in[2];
    const float* cvar_limit = (const float*)d_in[3];

    char* ws = (char*)d_ws;
    float* M        = (float*)ws;                                   // 268435456 B
    float* wB       = (float*)(ws + 268435456ULL);                  //   2097152 B
    float* partials = (float*)(ws + 268435456ULL + 2097152ULL);     //     16384 B
    int*   flag     = (int*)  (ws + 268435456ULL + 2097152ULL + 16384ULL);
    int*   counter  = flag + 1;

    float* wA = (float*)d_out;  // even-step buffer; 300 steps => result ends here

    init_kernel<<<(BH * NDIM + 255) / 256, 256, 0, stream>>>(wA, flag, counter);
    llt_gemm_kernel<<<BH, 256, 0, stream>>>(L, M);

    for (int it = 0; it < MAX_ITERS_C; ++it) {
        const float* win  = (it & 1) ? wB : wA;
        float*       wout = (it & 1) ? wA : wB;
        md_step_kernel<<<BH, 128, 0, stream>>>(M, mu, w_prev, cvar_limit,
                                               win, wout, partials, flag, counter);
    }
}